// GraphBaseline_48464410968239
// MI455X (gfx1250) — compile-verified
//
#include <hip/hip_runtime.h>

typedef __attribute__((ext_vector_type(2))) float v2f;
typedef __attribute__((ext_vector_type(8))) float v8f;

#define HIDDEN 128
#define OUTC   16
#define NGRAPH 128

// ---------- degree / normalization ----------
__global__ void gcn_deg_init(float* __restrict__ deg, int n) {
    int i = blockIdx.x * blockDim.x + threadIdx.x;
    if (i < n) deg[i] = 1.0f;  // self-loop contributes 1 to every node's dst-degree
}

__global__ void gcn_deg_edges(const long long* __restrict__ dst, float* __restrict__ deg, int e) {
    int i = blockIdx.x * blockDim.x + threadIdx.x;
    if (i < e) atomicAdd(&deg[(int)dst[i]], 1.0f);
}

__global__ void gcn_deg_finalize(float* __restrict__ deg, int n) {
    int i = blockIdx.x * blockDim.x + threadIdx.x;
    if (i < n) deg[i] = rsqrtf(deg[i]);  // deg >= 1 always (self loops)
}

// ---------- broadcast bias into the aggregation accumulator ----------
__global__ void gcn_bias_init(const float* __restrict__ b, float* __restrict__ out, int total) {
    int i = blockIdx.x * blockDim.x + threadIdx.x;
    if (i < total) out[i] = b[i & (HIDDEN - 1)];
}

// ---------- fp32 WMMA GEMM: out[M,128] = (relu?)A[M,128] @ W[128,128] ----------
// grid.x = M/16 blocks, 256 threads (8 waves); wave w computes tile (blockIdx.x, w).
// RELU is a compile-time flag so the inner loop has no per-element selects.
template <bool RELU>
__global__ void __launch_bounds__(256)
gcn_gemm_wmma(const float* __restrict__ A, const float* __restrict__ W,
              float* __restrict__ out) {
    __shared__ float Wl[HIDDEN * HIDDEN];  // 64 KB of 320 KB/WGP
    int tid = threadIdx.x;

    // stage W into LDS (coalesced float4)
    const float4* Wv  = (const float4*)W;
    float4*       Wlv = (float4*)Wl;
    #pragma unroll
    for (int i = tid; i < HIDDEN * HIDDEN / 4; i += 256) Wlv[i] = Wv[i];
    __syncthreads();

    int wave = tid >> 5;
    int lane = tid & 31;
    int m0 = blockIdx.x * 16;
    int n0 = wave * 16;

    // A fragment (16x4 f32): lanes 0-15 hold M=lane, K=k0+{0,1}; lanes 16-31 hold M=lane-16, K=k0+{2,3}
    int mrow = m0 + (lane & 15);
    int koff = (lane < 16) ? 0 : 2;
    // B fragment (4x16): v0 = row k0+koff, v1 = row k0+koff+1, col striped across lanes
    int col  = n0 + (lane & 15);

    const float* arow = A + (size_t)mrow * HIDDEN;
    v8f acc = {};
    #pragma unroll
    for (int k0 = 0; k0 < HIDDEN; k0 += 4) {
        v2f a;
        a.x = arow[k0 + koff];
        a.y = arow[k0 + koff + 1];
        if (RELU) {
            a.x = (a.x > 0.0f) ? a.x : 0.0f;
            a.y = (a.y > 0.0f) ? a.y : 0.0f;
        }
        v2f b;
        b.x = Wl[(k0 + koff) * HIDDEN + col];
        b.y = Wl[(k0 + koff + 1) * HIDDEN + col];
        // (neg_a, A, neg_b, B, c_mod, C, reuse_a, reuse_b)
        acc = __builtin_amdgcn_wmma_f32_16x16x4_f32(false, a, false, b,
                                                    (short)0, acc, false, false);
    }

    // C/D layout: VGPR v -> M = m0 + v (+8 for lanes 16-31), N = col
    int rbase = m0 + ((lane < 16) ? 0 : 8);
    #pragma unroll
    for (int v = 0; v < 8; ++v)
        out[(size_t)(rbase + v) * HIDDEN + col] = acc[v];
}

// ---------- edge aggregation: agg[dst] += h[src] * dis[src]*dis[dst] ----------
// one wave per edge (32 lanes x float4 = full 128-feature row); items >= n_edges are self-loops
__global__ void __launch_bounds__(256)
gcn_aggregate(const float* __restrict__ h, const long long* __restrict__ src,
              const long long* __restrict__ dst, const float* __restrict__ dis,
              float* __restrict__ agg, int n_edges, int n_nodes) {
    int item = blockIdx.x * 8 + (threadIdx.x >> 5);
    int lane = threadIdx.x & 31;
    if (item >= n_edges + n_nodes) return;
    int s, d;
    if (item < n_edges) { s = (int)src[item]; d = (int)dst[item]; }
    else                { s = d = item - n_edges; }
    float norm = dis[s] * dis[d];
    const float4* hv = (const float4*)(h + (size_t)s * HIDDEN);
    float4 v = hv[lane];
    float* ap = agg + (size_t)d * HIDDEN + lane * 4;
    atomicAdd(ap + 0, v.x * norm);
    atomicAdd(ap + 1, v.y * norm);
    atomicAdd(ap + 2, v.z * norm);
    atomicAdd(ap + 3, v.w * norm);
}

// ---------- pooling ----------
__global__ void gcn_pool_init(float* __restrict__ sums, float* __restrict__ counts) {
    int i = blockIdx.x * blockDim.x + threadIdx.x;
    if (i < NGRAPH * HIDDEN) sums[i] = 0.0f;
    if (i < NGRAPH) counts[i] = 0.0f;
}

__global__ void __launch_bounds__(256)
gcn_pool(const float* __restrict__ agg, const long long* __restrict__ batch,
         float* __restrict__ sums, float* __restrict__ counts, int n) {
    int node = blockIdx.x * 8 + (threadIdx.x >> 5);
    int lane = threadIdx.x & 31;
    if (node >= n) return;
    int g = (int)batch[node];
    const float4* hv = (const float4*)(agg + (size_t)node * HIDDEN);
    float4 v = hv[lane];  // fuse relu of layer-2 output here
    float* sp = sums + g * HIDDEN + lane * 4;
    atomicAdd(sp + 0, (v.x > 0.0f) ? v.x : 0.0f);
    atomicAdd(sp + 1, (v.y > 0.0f) ? v.y : 0.0f);
    atomicAdd(sp + 2, (v.z > 0.0f) ? v.z : 0.0f);
    atomicAdd(sp + 3, (v.w > 0.0f) ? v.w : 0.0f);
    if (lane == 0) atomicAdd(&counts[g], 1.0f);
}

// ---------- head: pooled @ Wfc + bfc, log_softmax; one thread per graph ----------
__global__ void __launch_bounds__(128)
gcn_head(const float* __restrict__ sums, const float* __restrict__ counts,
         const float* __restrict__ Wfc, const float* __restrict__ bfc,
         float* __restrict__ out) {
    int g = threadIdx.x;
    if (g >= NGRAPH) return;
    float inv = 1.0f / fmaxf(counts[g], 1.0f);
    float logits[OUTC];
    #pragma unroll
    for (int o = 0; o < OUTC; ++o) logits[o] = bfc[o];
    for (int c = 0; c < HIDDEN; ++c) {
        float p = sums[g * HIDDEN + c] * inv;
        #pragma unroll
        for (int o = 0; o < OUTC; ++o) logits[o] = fmaf(p, Wfc[c * OUTC + o], logits[o]);
    }
    float m = logits[0];
    #pragma unroll
    for (int o = 1; o < OUTC; ++o) m = fmaxf(m, logits[o]);
    float s = 0.0f;
    #pragma unroll
    for (int o = 0; o < OUTC; ++o) s += __expf(logits[o] - m);
    float ls = __logf(s);
    #pragma unroll
    for (int o = 0; o < OUTC; ++o) out[g * OUTC + o] = logits[o] - m - ls;
}

static inline size_t align256(size_t x) { return (x + 255) & ~(size_t)255; }

extern "C" void kernel_launch(void* const* d_in, const int* in_sizes, int n_in,
                              void* d_out, int out_size, void* d_ws, size_t ws_size,
                              hipStream_t stream) {
    const float*     x     = (const float*)d_in[0];
    const long long* edge  = (const long long*)d_in[1];  // int64 [2, E] row-major
    const long long* batch = (const long long*)d_in[2];  // int64 [N]
    const float*     W1    = (const float*)d_in[3];
    const float*     b1    = (const float*)d_in[4];
    const float*     W2    = (const float*)d_in[5];
    const float*     b2    = (const float*)d_in[6];
    const float*     Wfc   = (const float*)d_in[7];
    const float*     bfc   = (const float*)d_in[8];

    const int N = in_sizes[0] / HIDDEN;   // 50000
    const int E = in_sizes[1] / 2;        // 800000
    const long long* src = edge;
    const long long* dst = edge + E;

    // workspace carve-up
    char* ws = (char*)d_ws;
    size_t off = 0;
    float* dis = (float*)(ws + off); off += align256((size_t)N * 4);
    float* h   = (float*)(ws + off); off += align256((size_t)N * HIDDEN * 4);
    float* agg = (float*)(ws + off); off += align256((size_t)N * HIDDEN * 4);
    float* sums   = (float*)(ws + off); off += align256((size_t)NGRAPH * HIDDEN * 4);
    float* counts = (float*)(ws + off);

    const int T = 256;
    // 1. symmetric norm: dis = rsqrt(deg) with self-loops
    gcn_deg_init<<<(N + T - 1) / T, T, 0, stream>>>(dis, N);
    gcn_deg_edges<<<(E + T - 1) / T, T, 0, stream>>>(dst, dis, E);
    gcn_deg_finalize<<<(N + T - 1) / T, T, 0, stream>>>(dis, N);

    int items = E + N;
    int aggBlocks = (items + 7) / 8;

    // 2. layer 1: h = x @ W1 ; agg = b1 + scatter(norm * h[src])
    gcn_gemm_wmma<false><<<N / 16, T, 0, stream>>>(x, W1, h);
    gcn_bias_init<<<(N * HIDDEN + T - 1) / T, T, 0, stream>>>(b1, agg, N * HIDDEN);
    gcn_aggregate<<<aggBlocks, T, 0, stream>>>(h, src, dst, dis, agg, E, N);

    // 3. layer 2: h = relu(agg) @ W2 ; agg = b2 + scatter(norm * h[src])
    gcn_gemm_wmma<true><<<N / 16, T, 0, stream>>>(agg, W2, h);
    gcn_bias_init<<<(N * HIDDEN + T - 1) / T, T, 0, stream>>>(b2, agg, N * HIDDEN);
    gcn_aggregate<<<aggBlocks, T, 0, stream>>>(h, src, dst, dis, agg, E, N);

    // 4. mean pool (relu fused) + head
    gcn_pool_init<<<(NGRAPH * HIDDEN + T - 1) / T, T, 0, stream>>>(sums, counts);
    gcn_pool<<<(N + 7) / 8, T, 0, stream>>>(agg, batch, sums, counts, N);
    gcn_head<<<1, 128, 0, stream>>>(sums, counts, Wfc, bfc, (float*)d_out);
}